// ChairMLP_6657199309423
// MI455X (gfx1250) — compile-verified
//
#include <hip/hip_runtime.h>
#include <cstddef>

typedef float v2f __attribute__((ext_vector_type(2)));
typedef float v8f __attribute__((ext_vector_type(8)));

#define NPTS    262144
#define HDIM    60
#define SROW    68                 // state row stride (floats): bank-conflict-free, 8B aligned
#define WKP     160                // weight k-pair row stride (floats): halves hit disjoint banks
#define WLDSZ   (3 * 32 * WKP)     // 15360 floats
#define BLDSZ   (3 * 64)           // 192 floats
#define STATESZ (4 * 5 * 16 * SROW)// 21760 floats (4 waves x 5 comps x 16 pts x SROW)
#define SMEM_FLOATS (WLDSZ + BLDSZ + STATESZ)

// D = A(16x64) @ WT(64x64) via 16 k-steps of V_WMMA_F32_16X16X4_F32.
// A frag (16x4 f32): lane m=lane&15 holds row m; half hi=lane>>4 holds K=2*hi,2*hi+1 -> v2f.
// B frag (4x16 f32): lane holds col n=lane&15; half holds K=2*hi,2*hi+1 -> v2f
//   (weights staged pair-interleaved so both elements are one 8B LDS load).
// C/D (16x16 f32): VGPR v, lanes 0-15 -> row v, lanes 16-31 -> row v+8 (ISA 7.12.2).

__global__ __launch_bounds__(128, 1) void chair_mlp_wmma(
    const float* __restrict__ x,
    const float* __restrict__ w1, const float* __restrict__ b1,
    const float* __restrict__ w2, const float* __restrict__ b2,
    const float* __restrict__ w3, const float* __restrict__ b3,
    const float* __restrict__ w4, const float* __restrict__ b4,
    const float* __restrict__ w5, const float* __restrict__ b5,
    float* __restrict__ out)
{
    extern __shared__ float smem[];
    float* wlds = smem;                 // staged WT for layers 2..4, WMMA-B layout, zero-padded to 64x64
    float* blds = smem + WLDSZ;         // padded biases b2..b4
    float* sall = smem + WLDSZ + BLDSZ; // per-wave state [wave][comp(5)][pt(16)][SROW]

    const int tid  = threadIdx.x;
    const int lane = tid & 31;
    const int wid  = tid >> 5;

    // ---- stage padded weights/biases: wlds[l][kp][n][e] = WT[2kp+e][n] = w[n][2kp+e] ----
#pragma unroll
    for (int l = 0; l < 3; ++l) {
        const float* w = (l == 0) ? w2 : (l == 1) ? w3 : w4;
        const float* b = (l == 0) ? b2 : (l == 1) ? b3 : b4;
        float* wl = wlds + l * 32 * WKP;
        for (int f = tid; f < 4096; f += 128) {
            int e  = f & 1;
            int n  = (f >> 1) & 63;
            int kp = f >> 7;
            int k  = kp * 2 + e;
            wl[kp * WKP + n * 2 + e] = (k < HDIM && n < HDIM) ? w[n * HDIM + k] : 0.0f;
        }
        for (int i = tid; i < 64; i += 128)
            blds[l * 64 + i] = (i < HDIM) ? b[i] : 0.0f;
    }
    __syncthreads();

    const int p0 = (blockIdx.x * 4 + wid) * 16;   // this wave's first point
    float* S = sall + wid * (5 * 16 * SROW);      // comp r at S + r*16*SROW
    const float b5v = b5[0];

    // ---- layer 1 (3 -> 60), VALU; also emits alk1 ----
    for (int t = 0; t < 32; ++t) {
        int idx = lane + 32 * t;      // 16 pts x 64 cols
        int p = idx >> 6;
        int j = idx & 63;
        float h = 0.0f, q0 = 0.0f, q1 = 0.0f, q2 = 0.0f, qb = 0.0f;
        if (j < HDIM) {
            const float* xp = x + (size_t)(p0 + p) * 3;
            q0 = w1[j * 3 + 0]; q1 = w1[j * 3 + 1]; q2 = w1[j * 3 + 2]; qb = b1[j];
            h = fmaf(xp[0], q0, fmaf(xp[1], q1, fmaf(xp[2], q2, qb)));
            h = fmaxf(h, 0.0f);
        }
        float s  = (h > 0.0f) ? 1.0f : 0.0f;
        float c0 = q0 * s, c1 = q1 * s, c2 = q2 * s, cb = qb * s;
        S[0 * 16 * SROW + p * SROW + j] = h;
        S[1 * 16 * SROW + p * SROW + j] = c0;
        S[2 * 16 * SROW + p * SROW + j] = c1;
        S[3 * 16 * SROW + p * SROW + j] = c2;
        S[4 * 16 * SROW + p * SROW + j] = cb;
        if (j < HDIM) {
            float* a1 = out + NPTS + (size_t)(p0 + p) * 240;
            a1[0 * HDIM + j] = c0;
            a1[1 * HDIM + j] = c1;
            a1[2 * HDIM + j] = c2;
            a1[3 * HDIM + j] = cb;
        }
    }

    // ---- layers 2..4: five 16x64 @ 64x64 GEMMs per wave via fp32 WMMA ----
    const int m  = lane & 15;
    const int hi = lane >> 4;
    v8f hnew[4];
    const v8f vzero = {0, 0, 0, 0, 0, 0, 0, 0};

    for (int l = 0; l < 3; ++l) {
        const float* wl  = wlds + l * 32 * WKP;
        const float* bl  = blds + l * 64;
        float*       alk = out + NPTS + (size_t)(l + 1) * NPTS * 240;

        for (int r = 0; r < 5; ++r) {          // r: 0=h, 1..3=Wc rows, 4=bc
            float* Sr = S + r * 16 * SROW;

            v2f a[16];
#pragma unroll
            for (int k = 0; k < 16; ++k)
                a[k] = *(const v2f*)(Sr + m * SROW + 4 * k + 2 * hi);

            v8f d0 = vzero, d1 = vzero, d2 = vzero, d3 = vzero;
#pragma unroll
            for (int k = 0; k < 16; ++k) {
                const float* wk = wl + (2 * k + hi) * WKP + m * 2;
                v2f bf0 = *(const v2f*)(wk + 0);
                v2f bf1 = *(const v2f*)(wk + 32);
                v2f bf2 = *(const v2f*)(wk + 64);
                v2f bf3 = *(const v2f*)(wk + 96);
                d0 = __builtin_amdgcn_wmma_f32_16x16x4_f32(false, a[k], false, bf0, (short)0, d0, false, false);
                d1 = __builtin_amdgcn_wmma_f32_16x16x4_f32(false, a[k], false, bf1, (short)0, d1, false, false);
                d2 = __builtin_amdgcn_wmma_f32_16x16x4_f32(false, a[k], false, bf2, (short)0, d2, false, false);
                d3 = __builtin_amdgcn_wmma_f32_16x16x4_f32(false, a[k], false, bf3, (short)0, d3, false, false);
            }
            v8f dd[4] = {d0, d1, d2, d3};

            if (r == 0) {                       // h: bias + relu; keep mask fragments
#pragma unroll
                for (int ct = 0; ct < 4; ++ct) {
                    int col = ct * 16 + m;
                    float bias = bl[col];
#pragma unroll
                    for (int v = 0; v < 8; ++v) {
                        float hv = fmaxf(dd[ct][v] + bias, 0.0f);
                        hnew[ct][v] = hv;
                        Sr[(v + 8 * hi) * SROW + col] = hv;
                    }
                }
            } else {                            // affine rows: (+b for bc) then mask by s
#pragma unroll
                for (int ct = 0; ct < 4; ++ct) {
                    int col = ct * 16 + m;
                    float bias = (r == 4) ? bl[col] : 0.0f;
#pragma unroll
                    for (int v = 0; v < 8; ++v) {
                        float val = dd[ct][v] + bias;
                        val = (hnew[ct][v] > 0.0f) ? val : 0.0f;
                        Sr[(v + 8 * hi) * SROW + col] = val;
                        if (col < HDIM)
                            alk[(size_t)(p0 + v + 8 * hi) * 240 + (size_t)(r - 1) * HDIM + col] = val;
                    }
                }
            }
        }
    }

    // ---- layer 5 (60 -> 1) + zerosurface, VALU dots from LDS state ----
    float* zs = out + NPTS + (size_t)4 * NPTS * 240;
    for (int t = 0; t < 3; ++t) {
        int task = lane + 32 * t;               // 16 pts x 5 comps = 80 dots
        if (task < 80) {
            int p = task & 15;
            int c = task >> 4;                  // 0=h->out, 1..3=Wc->zs, 4=bc->zs
            const float* Sr = S + c * 16 * SROW + p * SROW;
            float acc = 0.0f;
            for (int k = 0; k < HDIM; ++k)
                acc = fmaf(Sr[k], w5[k], acc);
            int gp = p0 + p;
            if (c == 0)      out[gp] = acc + b5v;
            else if (c < 4)  zs[(size_t)gp * 4 + (c - 1)] = acc;
            else             zs[(size_t)gp * 4 + 3] = acc + b5v;
        }
    }
}

extern "C" void kernel_launch(void* const* d_in, const int* in_sizes, int n_in,
                              void* d_out, int out_size, void* d_ws, size_t ws_size,
                              hipStream_t stream) {
    const float* x  = (const float*)d_in[0];
    const float* w1 = (const float*)d_in[1];
    const float* b1 = (const float*)d_in[2];
    const float* w2 = (const float*)d_in[3];
    const float* b2 = (const float*)d_in[4];
    const float* w3 = (const float*)d_in[5];
    const float* b3 = (const float*)d_in[6];
    const float* w4 = (const float*)d_in[7];
    const float* b4 = (const float*)d_in[8];
    const float* w5 = (const float*)d_in[9];
    const float* b5 = (const float*)d_in[10];
    float* out = (float*)d_out;

    size_t shmem = (size_t)SMEM_FLOATS * sizeof(float);  // ~146 KB dynamic LDS
    dim3 grid(NPTS / 64), block(128);                    // 4 waves x 16 points = 64 pts/WG
    chair_mlp_wmma<<<grid, block, shmem, stream>>>(x, w1, b1, w2, b2, w3, b3, w4, b4, w5, b5, out);

    (void)in_sizes; (void)n_in; (void)out_size; (void)d_ws; (void)ws_size;
}